// GCNConv_G_86148454023366
// MI455X (gfx1250) — compile-verified
//
#include <hip/hip_runtime.h>
#include <stdint.h>

typedef __attribute__((ext_vector_type(2))) float v2f;
typedef __attribute__((ext_vector_type(8))) float v8f;

#define EPS_NUM 1e-6f

__device__ __forceinline__ unsigned f2ord(float f) {
  unsigned u = __float_as_uint(f);
  return (u & 0x80000000u) ? ~u : (u | 0x80000000u);
}
__device__ __forceinline__ float ord2f(unsigned k) {
  return (k & 0x80000000u) ? __uint_as_float(k & 0x7fffffffu)
                           : __uint_as_float(~k);
}
__device__ __forceinline__ void atomAddF(float* p, float v) {
  __hip_atomic_fetch_add(p, v, __ATOMIC_RELAXED, __HIP_MEMORY_SCOPE_AGENT);
}

// ---------------- global min of x -> ordered-uint atomicMin ----------------
__global__ void kmin(const float4* __restrict__ x, long long n4,
                     unsigned* __restrict__ mukey) {
  long long i = blockIdx.x * (long long)blockDim.x + threadIdx.x;
  long long stride = (long long)gridDim.x * blockDim.x;
  float m = __builtin_inff();
  for (; i < n4; i += stride) {
    float4 v = x[i];
    m = fminf(m, fminf(fminf(v.x, v.y), fminf(v.z, v.w)));
  }
#pragma unroll
  for (int off = 16; off > 0; off >>= 1)
    m = fminf(m, __shfl_xor(m, off, 32));
  if ((threadIdx.x & 31) == 0) atomicMin(mukey, f2ord(m));
}

// ---------------- in-degree of col via native f32 atomics ----------------
__global__ void kdeg(const int* __restrict__ col, int E, float* __restrict__ deg) {
  int e = blockIdx.x * blockDim.x + threadIdx.x;
  if (e < E) atomAddF(&deg[(unsigned)col[e]], 1.0f);  // saddr-form atomic
}

// ---------------- deg -> deg^{-1/2} (0 if deg==0), in place ----------------
__global__ void kdinv(float* __restrict__ deg, int N) {
  int i = blockIdx.x * blockDim.x + threadIdx.x;
  if (i < N) {
    float d = deg[i];
    deg[i] = (d > 0.0f) ? __frsqrt_rn(d) : 0.0f;
  }
}

// ---------------- x1 = (x - mu + eps)^pp ----------------
__global__ void kpow(const float4* __restrict__ x, float4* __restrict__ x1,
                     const unsigned* __restrict__ mukey,
                     const float* __restrict__ pvec, long long n4) {
  float mu = ord2f(*mukey);
  float pp = 1.0f / (1.0f + expf(-pvec[0])) + 1.0f;
  long long i = blockIdx.x * (long long)blockDim.x + threadIdx.x;
  if (i >= n4) return;
  float4 v = x[i];
  float4 o;
  o.x = exp2f(pp * log2f(v.x - mu + EPS_NUM));
  o.y = exp2f(pp * log2f(v.y - mu + EPS_NUM));
  o.z = exp2f(pp * log2f(v.z - mu + EPS_NUM));
  o.w = exp2f(pp * log2f(v.w - mu + EPS_NUM));
  x1[i] = o;
}

// ---- SpMM helpers: async-stage a 16-edge tile, compute+scatter a tile ----

// Issue 16 async B128 DMA loads staging the gathered rows of tile t into LDS.
// saddr form: uniform x1 base in SGPRs + 32-bit per-lane byte offset.
__device__ __forceinline__ void stage_tile(
    const float* __restrict__ x1, const int* __restrict__ row,
    const int* __restrict__ col, const float* __restrict__ dinv, int E, int t,
    unsigned ldsBase, int lane, int& r_out, float& nrm_out) {
  const int m = lane & 15;
  int e = (t << 4) + m;
  bool valid = e < E;
  int ec = valid ? e : 0;
  int r = row[ec];
  int c = col[ec];
  float nrm = valid ? dinv[(unsigned)r] * dinv[(unsigned)c] : 0.0f;
  const unsigned lane16 = (unsigned)lane << 4;  // 16B per lane
#pragma unroll
  for (int j = 0; j < 16; ++j) {
    unsigned cj = (unsigned)__shfl(c, j, 32);
    unsigned goff = (cj << 9) + lane16;                   // byte offset in x1
    unsigned lp = ldsBase + (unsigned)j * 544u + lane16;  // LDS dest
    asm volatile("global_load_async_to_lds_b128 %0, %1, %2"
                 :: "v"(lp), "v"(goff), "s"(x1) : "memory");
  }
  r_out = r;
  nrm_out = nrm;
}

// WMMA diag(norm) x LDS-tile, scatter-add via saddr atomics (32-bit offsets).
__device__ __forceinline__ void compute_tile(const float (*tb)[136],
                                             float* __restrict__ agg, int lane,
                                             int r, float nrm) {
  const int m = lane & 15;
  const int kh = lane >> 4;
  unsigned voff[8];
#pragma unroll
  for (int v = 0; v < 8; ++v) {  // my C/D rows: M = v + 8*kh
    unsigned rv = (unsigned)__shfl(r, v + (kh << 3), 32);
    voff[v] = (rv << 9) + ((unsigned)m << 2);  // byte offset into agg
  }
  v2f a[4];
#pragma unroll
  for (int ck = 0; ck < 4; ++ck) {  // diag(norm) K-slices
    int k0 = (ck << 2) + (kh << 1);
    a[ck].x = (m == k0) ? nrm : 0.0f;
    a[ck].y = (m == k0 + 1) ? nrm : 0.0f;
  }
#pragma unroll
  for (int ft = 0; ft < 8; ++ft) {  // 8 feature tiles of 16 (fully unrolled)
    v8f acc = {};
#pragma unroll
    for (int ck = 0; ck < 4; ++ck) {  // K=16 as 4 chained K=4 WMMAs
      int k0 = (ck << 2) + (kh << 1);
      v2f b;
      b.x = tb[k0][ft * 16 + m];
      b.y = tb[k0 + 1][ft * 16 + m];
      acc = __builtin_amdgcn_wmma_f32_16x16x4_f32(false, a[ck], false, b,
                                                  (short)0, acc, false, false);
    }
#pragma unroll
    for (int v = 0; v < 8; ++v)  // ft*64B folds into the atomic's imm offset
      atomAddF((float*)((char*)agg + voff[v] + (unsigned)(ft << 6)), acc[v]);
  }
}

// Persistent waves, double-buffered LDS: DMA of tile t+1 overlaps WMMA+atomics
// of tile t. Async loads complete in order, so asynccnt<=16 after issuing the
// next 16 guarantees the current tile is resident.
__global__ __launch_bounds__(128) void kspmm(
    const float* __restrict__ x1, const float* __restrict__ dinv,
    const int* __restrict__ row, const int* __restrict__ col,
    float* __restrict__ agg, int E) {
  __shared__ float tile[4][2][16][136];  // 544B row stride, 2 bufs per wave
  const int lane = threadIdx.x & 31;
  const int wv = threadIdx.x >> 5;
  const int tiles = (E + 15) >> 4;
  const int stride = gridDim.x * 4;
  int t = blockIdx.x * 4 + wv;  // uniform per wave
  if (t >= tiles) return;       // whole-wave exit: EXEC stays all-ones

  unsigned lb[2];
  lb[0] = (unsigned)(unsigned long long)(uintptr_t)&tile[wv][0][0][0];
  lb[1] = (unsigned)(unsigned long long)(uintptr_t)&tile[wv][1][0][0];

  int r_cur;
  float nrm_cur;
  stage_tile(x1, row, col, dinv, E, t, lb[0], lane, r_cur, nrm_cur);
  int buf = 0;
  for (;;) {
    int tn = t + stride;
    bool hasNext = tn < tiles;  // wave-uniform
    int r_nxt = 0;
    float nrm_nxt = 0.0f;
    if (hasNext) {
      stage_tile(x1, row, col, dinv, E, tn, lb[buf ^ 1], lane, r_nxt, nrm_nxt);
      asm volatile("s_wait_asynccnt 0x10" ::: "memory");  // cur tile resident
    } else {
      asm volatile("s_wait_asynccnt 0x0" ::: "memory");
    }
    compute_tile((const float(*)[136])tile[wv][buf], agg, lane, r_cur, nrm_cur);
    if (!hasNext) break;
    t = tn;
    r_cur = r_nxt;
    nrm_cur = nrm_nxt;
    buf ^= 1;
  }
}

// ---------------- out = (agg + eps)^(1/pp) + (1+eps0)*x + mu ----------------
__global__ void kfinal(const float4* __restrict__ agg, const float4* __restrict__ x,
                       float4* __restrict__ out, const unsigned* __restrict__ mukey,
                       const float* __restrict__ pvec,
                       const float* __restrict__ epsv, long long n4) {
  float mu = ord2f(*mukey);
  float pp = 1.0f / (1.0f + expf(-pvec[0])) + 1.0f;
  float ipp = 1.0f / pp;
  float se = 1.0f + epsv[0];
  long long i = blockIdx.x * (long long)blockDim.x + threadIdx.x;
  if (i >= n4) return;
  float4 a = agg[i];
  float4 v = x[i];
  float4 o;
  o.x = exp2f(ipp * log2f(a.x + EPS_NUM)) + se * v.x + mu;
  o.y = exp2f(ipp * log2f(a.y + EPS_NUM)) + se * v.y + mu;
  o.z = exp2f(ipp * log2f(a.z + EPS_NUM)) + se * v.z + mu;
  o.w = exp2f(ipp * log2f(a.w + EPS_NUM)) + se * v.w + mu;
  out[i] = o;
}

extern "C" void kernel_launch(void* const* d_in, const int* in_sizes, int n_in,
                              void* d_out, int out_size, void* d_ws, size_t ws_size,
                              hipStream_t stream) {
  const float* x = (const float*)d_in[0];
  const float* epsv = (const float*)d_in[1];
  const float* pv = (const float*)d_in[2];
  const int* ei = (const int*)d_in[3];
  float* out = (float*)d_out;

  const long long NE = in_sizes[0];  // N * 128
  const int N = (int)(NE / 128);
  const int E = in_sizes[3] / 2;
  const int* row = ei;
  const int* col = ei + E;

  char* w = (char*)d_ws;
  unsigned* mukey = (unsigned*)w;
  size_t degB = (((size_t)N * 4) + 255) & ~(size_t)255;
  float* deg = (float*)(w + 256);
  float* x1 = (float*)(w + 256 + degB);
  float* agg = x1 + (size_t)N * 128;

  hipMemsetAsync(mukey, 0xFF, 4, stream);  // +inf ordered key
  hipMemsetAsync(deg, 0, (size_t)N * 4, stream);
  hipMemsetAsync(agg, 0, (size_t)N * 512, stream);

  long long n4 = NE / 4;
  kmin<<<1024, 256, 0, stream>>>((const float4*)x, n4, mukey);
  kdeg<<<(E + 255) / 256, 256, 0, stream>>>(col, E, deg);
  kdinv<<<(N + 255) / 256, 256, 0, stream>>>(deg, N);
  kpow<<<(unsigned)((n4 + 255) / 256), 256, 0, stream>>>(
      (const float4*)x, (float4*)x1, mukey, pv, n4);

  int tiles = (E + 15) / 16;
  int blocks = (tiles + 3) / 4;
  if (blocks > 2048) blocks = 2048;  // persistent, grid-stride over tiles
  kspmm<<<blocks, 128, 0, stream>>>(x1, deg, row, col, agg, E);

  kfinal<<<(unsigned)((n4 + 255) / 256), 256, 0, stream>>>(
      (const float4*)agg, (const float4*)x, (float4*)out, mukey, pv, epsv, n4);
}